// PointerNetworkVRPTW_39479339385491
// MI455X (gfx1250) — compile-verified
//
#include <hip/hip_runtime.h>
#include <hip/hip_bf16.h>
#include <stdint.h>

#define HID 128
#define BATCH 256
#define SEQ 512

typedef float v2f __attribute__((ext_vector_type(2)));
typedef float v8f __attribute__((ext_vector_type(8)));

// ---------------- threefry2x32 (bit-exact JAX) ----------------
__device__ __forceinline__ uint32_t rotl32(uint32_t x, uint32_t n) {
  return (x << n) | (x >> (32u - n));
}

__device__ __forceinline__ void threefry2x32(uint32_t k0, uint32_t k1,
                                             uint32_t c0, uint32_t c1,
                                             uint32_t& o0, uint32_t& o1) {
  uint32_t ks0 = k0, ks1 = k1, ks2 = k0 ^ k1 ^ 0x1BD11BDAu;
  uint32_t x0 = c0 + ks0, x1 = c1 + ks1;
#define TF_R(r) { x0 += x1; x1 = rotl32(x1, r); x1 ^= x0; }
  TF_R(13u) TF_R(15u) TF_R(26u) TF_R(6u)  x0 += ks1; x1 += ks2 + 1u;
  TF_R(17u) TF_R(29u) TF_R(16u) TF_R(24u) x0 += ks2; x1 += ks0 + 2u;
  TF_R(13u) TF_R(15u) TF_R(26u) TF_R(6u)  x0 += ks0; x1 += ks1 + 3u;
  TF_R(17u) TF_R(29u) TF_R(16u) TF_R(24u) x0 += ks1; x1 += ks2 + 4u;
  TF_R(13u) TF_R(15u) TF_R(26u) TF_R(6u)  x0 += ks2; x1 += ks0 + 5u;
#undef TF_R
  o0 = x0; o1 = x1;
}

__device__ __forceinline__ float gumbel_from_bits(uint32_t bits) {
  // JAX uniform: bitcast((bits>>9)|0x3f800000)-1 in [0,1), then [tiny,1)
  uint32_t m = (bits >> 9) | 0x3f800000u;
  float f = __uint_as_float(m) - 1.0f;
  const float tiny = 1.1754944e-38f;
  float u = f * (1.0f - tiny) + tiny;
  return -logf(-logf(u));
}

// ---------------- prep: wdwa/bdwa, weight transposes, split keys ----------------
__global__ void __launch_bounds__(256) prep_kernel(
    const float* __restrict__ Wd, const float* __restrict__ bd,
    const float* __restrict__ Wa_d, const float* __restrict__ W_ih,
    const float* __restrict__ W_hh, float* __restrict__ wdwa,
    float* __restrict__ bdwa, float* __restrict__ WihT,
    float* __restrict__ WhhT, uint32_t* __restrict__ keysG) {
  __shared__ uint32_t outL[1024];
  int tid = threadIdx.x;
  // jax.random.split(key(42), 512): threefry((0,42), iota(1024))
  for (int L = tid; L < 512; L += 256) {
    uint32_t o0, o1;
    threefry2x32(0u, 42u, (uint32_t)L, (uint32_t)(L + 512), o0, o1);
    outL[L] = o0; outL[L + 512] = o1;
  }
  __syncthreads();
  for (int j = tid; j < 1024; j += 256) keysG[j] = outL[j];
  // wdwa = Wd @ Wa_d ; bdwa = bd @ Wa_d  (rank-1 collapse of de@Wa_d)
  if (tid < HID) {
    float aw = 0.f, ab = 0.f;
    for (int j = 0; j < HID; ++j) {
      float w = Wa_d[j * HID + tid];
      aw += Wd[j] * w;
      ab += bd[j] * w;
    }
    wdwa[tid] = aw; bdwa[tid] = ab;
  }
  // transpose GRU weights for coalesced GEMV: WT[k*384+j] = W[j*128+k]
  for (int e = tid; e < 384 * HID; e += 256) {
    int k = e / 384, j = e - k * 384;
    WihT[e] = W_ih[j * HID + k];
    WhhT[e] = W_hh[j * HID + k];
  }
}

// ---------------- se + proj_s GEMM via v_wmma_f32_16x16x4_f32 ----------------
// proj2[R][n] = (se @ Wa_s)[R][n] + bdwa[n], R = b*SEQ + s; also dec0 = se[:,0,:]
__global__ void __launch_bounds__(256) seproj_kernel(
    const float* __restrict__ statc, const float* __restrict__ Ws,
    const float* __restrict__ bs, const float* __restrict__ Wa_s,
    const float* __restrict__ bdwa, float* __restrict__ proj2,
    float* __restrict__ dec0) {
  __shared__ float seL[128 * HID];  // 64 KB tile of se rows
  int tid = threadIdx.x, lane = tid & 31, w = tid >> 5;
  int rowBase = blockIdx.x * 128;
  // se[r][h] = bs[h] + sum_f static[b,f,s]*Ws[f,h]
  for (int e = tid; e < 128 * HID; e += 256) {
    int r = e >> 7, hh = e & 127;
    int R = rowBase + r, b = R >> 9, s = R & 511;
    float acc = bs[hh];
#pragma unroll
    for (int f = 0; f < 4; ++f)
      acc += statc[((size_t)(b * 4 + f)) * SEQ + s] * Ws[f * HID + hh];
    seL[e] = acc;
    if (s == 0) dec0[b * HID + hh] = acc;  // initial dec_in = se[:,0,:]
  }
  __syncthreads();
  // wave w computes rows [w*16, w*16+16) x all 128 cols as 8 WMMA tiles
  int rw = w * 16;
  v8f acc[8];
#pragma unroll
  for (int nt = 0; nt < 8; ++nt) acc[nt] = (v8f){0, 0, 0, 0, 0, 0, 0, 0};
  int mrow = rw + (lane & 15);
  int khalf = (lane >> 4) << 1;  // lanes 16-31 carry K+2,K+3
  int ncol = lane & 15;
  for (int k = 0; k < HID; k += 4) {
    int kl = k + khalf;
    v2f a;
    a[0] = seL[mrow * HID + kl];
    a[1] = seL[mrow * HID + kl + 1];
#pragma unroll
    for (int nt = 0; nt < 8; ++nt) {
      v2f bf;
      bf[0] = Wa_s[kl * HID + nt * 16 + ncol];
      bf[1] = Wa_s[(kl + 1) * HID + nt * 16 + ncol];
      acc[nt] = __builtin_amdgcn_wmma_f32_16x16x4_f32(
          false, a, false, bf, (short)0, acc[nt], false, false);
    }
  }
  // C/D layout: VGPR v, lanes0-15 -> M=v, lanes16-31 -> M=v+8
  int rsel = rw + ((lane >> 4) << 3);
#pragma unroll
  for (int nt = 0; nt < 8; ++nt) {
    int col = nt * 16 + ncol;
    float badd = bdwa[col];
#pragma unroll
    for (int v = 0; v < 8; ++v) {
      int row = rsel + v;
      proj2[((size_t)(rowBase + row)) * HID + col] = acc[nt][v] + badd;
    }
  }
}

// ---------------- persistent per-batch scan (512 steps in-kernel) ----------------
__global__ void __launch_bounds__(256) scan_kernel(
    const float* __restrict__ dynamic, const float* __restrict__ dist,
    const float* __restrict__ Wi, const float* __restrict__ bi,
    const float* __restrict__ b_ih, const float* __restrict__ b_hh,
    const float* __restrict__ Wa_h, const float* __restrict__ v_att,
    const float* __restrict__ proj2, const float* __restrict__ dec0,
    const float* __restrict__ wdwa, const float* __restrict__ WihT,
    const float* __restrict__ WhhT, const uint32_t* __restrict__ keysG,
    float* __restrict__ outTours, float* __restrict__ outLogp,
    float* __restrict__ outTime) {
  __shared__ float dyn0L[SEQ], maskL[SEQ], logitsL[SEQ];
  __shared__ float giL[384], ghL[384], bihL[384], bhhL[384];
  __shared__ float xL[HID], decL[HID], hL[HID], hWaL[HID];
  __shared__ float biL[HID], vattL[HID], wdwaL[HID];
  __shared__ float redv[256];
  __shared__ int redi[256];
  __shared__ uint32_t keysL[1024];
  __shared__ float sCum;
  __shared__ int sOld;

  int tid = threadIdx.x, lane = tid & 31, w = tid >> 5;
  int b = blockIdx.x;

  for (int s = tid; s < SEQ; s += 256) {
    dyn0L[s] = dynamic[(size_t)b * SEQ + s];
    maskL[s] = 1.f;
  }
  if (tid < HID) {
    hL[tid] = 0.f;
    decL[tid] = dec0[b * HID + tid];
    biL[tid] = bi[tid];
    vattL[tid] = v_att[tid];
    wdwaL[tid] = wdwa[tid];
  }
  for (int j = tid; j < 384; j += 256) { bihL[j] = b_ih[j]; bhhL[j] = b_hh[j]; }
  for (int j = tid; j < 1024; j += 256) keysL[j] = keysG[j];
  if (tid == 0) { sCum = 0.f; sOld = 0; }
  __syncthreads();

  for (int t = 0; t < SEQ; ++t) {
    // x = dec_in @ Wi + bi
    if (tid < HID) {
      float a = biL[tid];
      for (int k = 0; k < HID; ++k) a += decL[k] * Wi[k * HID + tid];
      xL[tid] = a;
    }
    __syncthreads();
    // gi = x @ W_ih.T + b_ih ; gh = h @ W_hh.T + b_hh
    for (int out = tid; out < 768; out += 256) {
      if (out < 384) {
        float a = bihL[out];
        for (int k = 0; k < HID; ++k) a += xL[k] * WihT[k * 384 + out];
        giL[out] = a;
      } else {
        int o = out - 384;
        float a = bhhL[o];
        for (int k = 0; k < HID; ++k) a += hL[k] * WhhT[k * 384 + o];
        ghL[o] = a;
      }
    }
    __syncthreads();
    // GRU gates
    if (tid < HID) {
      float r = 1.f / (1.f + expf(-(giL[tid] + ghL[tid])));
      float z = 1.f / (1.f + expf(-(giL[HID + tid] + ghL[HID + tid])));
      float n = tanhf(giL[2 * HID + tid] + r * ghL[2 * HID + tid]);
      hL[tid] = (1.f - z) * n + z * hL[tid];
      decL[tid] = xL[tid];  // dec_in for next step = x
    }
    __syncthreads();
    // hWa = h @ Wa_h
    if (tid < HID) {
      float a = 0.f;
      for (int k = 0; k < HID; ++k) a += hL[k] * Wa_h[k * HID + tid];
      hWaL[tid] = a;
    }
    __syncthreads();
    // attention scores: u[s] = sum_h tanh(proj2 + dyn_s*wdwa + hWa) * v_att
    {
      float ct = sCum;
      const float4* base = (const float4*)(proj2 + ((size_t)b * SEQ) * HID);
      int hb = lane << 2;
      float wd0 = wdwaL[hb], wd1 = wdwaL[hb + 1], wd2 = wdwaL[hb + 2], wd3 = wdwaL[hb + 3];
      float hw0 = hWaL[hb], hw1 = hWaL[hb + 1], hw2 = hWaL[hb + 2], hw3 = hWaL[hb + 3];
      float va0 = vattL[hb], va1 = vattL[hb + 1], va2 = vattL[hb + 2], va3 = vattL[hb + 3];
      for (int it = 0; it < 64; ++it) {
        int s = (w << 6) + it;
        float4 p = base[s * 32 + lane];
        if (it < 63)
          __builtin_prefetch((const void*)(base + (s + 1) * 32 + lane), 0, 0);
        float ds_ = dyn0L[s] + ct;
        float val = tanhf(p.x + ds_ * wd0 + hw0) * va0 +
                    tanhf(p.y + ds_ * wd1 + hw1) * va1 +
                    tanhf(p.z + ds_ * wd2 + hw2) * va2 +
                    tanhf(p.w + ds_ * wd3 + hw3) * va3;
        for (int off = 16; off > 0; off >>= 1) val += __shfl_xor(val, off, 32);
        if (lane == 0) {
          float m = (t == 0) ? (s == 0 ? 1.f : 0.f) : maskL[s];
          logitsL[s] = (m != 0.f) ? val : -1.0e9f;
        }
      }
    }
    __syncthreads();
    // Gumbel sampling (bit-exact threefry) + log_softmax
    float l0 = logitsL[tid], l1 = logitsL[tid + 256];
    uint32_t k0 = keysL[2 * t], k1 = keysL[2 * t + 1];
    float g0, g1;
    {
      uint32_t i0 = (uint32_t)(b * SEQ + tid);
      uint32_t i1 = i0 + 256u;
      uint32_t o0, o1;
      if (b < 128) {  // flat index < 65536 -> first threefry output stream
        threefry2x32(k0, k1, i0, i0 + 65536u, o0, o1); g0 = gumbel_from_bits(o0);
        threefry2x32(k0, k1, i1, i1 + 65536u, o0, o1); g1 = gumbel_from_bits(o0);
      } else {
        threefry2x32(k0, k1, i0 - 65536u, i0, o0, o1); g0 = gumbel_from_bits(o1);
        threefry2x32(k0, k1, i1 - 65536u, i1, o0, o1); g1 = gumbel_from_bits(o1);
      }
    }
    float s0v = l0 + g0, s1v = l1 + g1;
    float bv; int bidx;
    if (s1v > s0v) { bv = s1v; bidx = tid + 256; } else { bv = s0v; bidx = tid; }
    redv[tid] = bv; redi[tid] = bidx;
    __syncthreads();
    for (int off = 128; off > 0; off >>= 1) {
      if (tid < off) {
        if (redv[tid + off] > redv[tid]) {
          redv[tid] = redv[tid + off];
          redi[tid] = redi[tid + off];
        }
      }
      __syncthreads();
    }
    // log-sum-exp over logits
    redv[tid] = fmaxf(l0, l1);
    __syncthreads();
    for (int off = 128; off > 0; off >>= 1) {
      if (tid < off) redv[tid] = fmaxf(redv[tid], redv[tid + off]);
      __syncthreads();
    }
    float maxl = redv[0];
    __syncthreads();
    redv[tid] = expf(l0 - maxl) + expf(l1 - maxl);
    __syncthreads();
    for (int off = 128; off > 0; off >>= 1) {
      if (tid < off) redv[tid] += redv[tid + off];
      __syncthreads();
    }
    if (tid == 0) {
      int p = redi[0];
      float lse = maxl + logf(redv[0]);
      outTours[(size_t)b * SEQ + t] = (float)p;
      outLogp[(size_t)b * SEQ + t] = logitsL[p] - lse;
      float tr = dist[(((size_t)b * SEQ + (size_t)sOld) * SEQ) + p];
      sCum = sCum + tr;
      sOld = p;
      maskL[p] = 0.f;
    }
    __syncthreads();
  }
  if (tid == 0) outTime[b] = dyn0L[0] + sCum;
}

// ---------------- host launch ----------------
extern "C" void kernel_launch(void* const* d_in, const int* in_sizes, int n_in,
                              void* d_out, int out_size, void* d_ws,
                              size_t ws_size, hipStream_t stream) {
  (void)in_sizes; (void)n_in; (void)out_size; (void)ws_size;
  const float* statc   = (const float*)d_in[0];
  const float* dynamic = (const float*)d_in[1];
  const float* dist    = (const float*)d_in[2];
  const float* Ws      = (const float*)d_in[3];
  const float* bs      = (const float*)d_in[4];
  const float* Wd      = (const float*)d_in[5];
  const float* bd      = (const float*)d_in[6];
  const float* Wi      = (const float*)d_in[7];
  const float* bi      = (const float*)d_in[8];
  const float* W_ih    = (const float*)d_in[9];
  const float* W_hh    = (const float*)d_in[10];
  const float* b_ih    = (const float*)d_in[11];
  const float* b_hh    = (const float*)d_in[12];
  const float* Wa_s    = (const float*)d_in[13];
  const float* Wa_d    = (const float*)d_in[14];
  const float* Wa_h    = (const float*)d_in[15];
  const float* v_att   = (const float*)d_in[16];

  // workspace layout (floats); proj2 (64MB) dominates and stays L2-resident
  float* ws = (float*)d_ws;
  size_t off = 0;
  float* proj2 = ws + off; off += (size_t)BATCH * SEQ * HID;  // 16,777,216
  float* dec0  = ws + off; off += (size_t)BATCH * HID;
  float* wdwa  = ws + off; off += HID;
  float* bdwa  = ws + off; off += HID;
  float* WihT  = ws + off; off += (size_t)384 * HID;
  float* WhhT  = ws + off; off += (size_t)384 * HID;
  uint32_t* keysG = (uint32_t*)(ws + off);

  float* outTours = (float*)d_out;
  float* outLogp  = outTours + (size_t)BATCH * SEQ;
  float* outTime  = outLogp + (size_t)BATCH * SEQ;

  prep_kernel<<<dim3(1), dim3(256), 0, stream>>>(
      Wd, bd, Wa_d, W_ih, W_hh, wdwa, bdwa, WihT, WhhT, keysG);
  seproj_kernel<<<dim3((BATCH * SEQ) / 128), dim3(256), 0, stream>>>(
      statc, Ws, bs, Wa_s, bdwa, proj2, dec0);
  scan_kernel<<<dim3(BATCH), dim3(256), 0, stream>>>(
      dynamic, dist, Wi, bi, b_ih, b_hh, Wa_h, v_att, proj2, dec0, wdwa, WihT,
      WhhT, keysG, outTours, outLogp, outTime);
}